// PositionTerm_82592221102151
// MI455X (gfx1250) — compile-verified
//
#include <hip/hip_runtime.h>
#include <hip/hip_bf16.h>
#include <math.h>

typedef __attribute__((ext_vector_type(2))) float v2f;
typedef __attribute__((ext_vector_type(4))) float v4f;
typedef __attribute__((ext_vector_type(8))) float v8f;

#define EMB_DIM 512   // per-axis sinusoid width
#define OUT_DIM 256
#define HW      64
#define BATCH   8

// ---------------------------------------------------------------------------
// Kernel 1: build the two sinusoid tables ex (64 x 512) and ey (64 x 512).
// embed(z)[d] = sin(z / 1000^((2d+1)/512))        for d in [0,256)
//             = cos(z / 1000^((2(d-256))/512))    for d in [256,512)
// 65536 total elements -> trivial VALU work.
// ---------------------------------------------------------------------------
__global__ void PositionTerm_embed_kernel(float* __restrict__ ex,
                                          float* __restrict__ ey) {
    const unsigned t = blockIdx.x * blockDim.x + threadIdx.x; // 0 .. 2*64*512-1
    const int d    = t & (EMB_DIM - 1);
    const int z    = (t >> 9) & (HW - 1);
    const int axis = t >> 15;               // 0 -> ex, 1 -> ey
    const float zf = (float)z;

    float e;
    if (d < EMB_DIM / 2) e = (2.0f * d + 1.0f) * (1.0f / EMB_DIM);
    else                 e = (2.0f * (d - EMB_DIM / 2)) * (1.0f / EMB_DIM);
    // 1000^(-e) = exp(-e * ln(1000))
    const float inv = expf(-e * 6.90775527898213705f);
    const float arg = zf * inv;
    const float val = (d < EMB_DIM / 2) ? sinf(arg) : cosf(arg);

    float* dst = axis ? ey : ex;
    dst[z * EMB_DIM + d] = val;
}

// ---------------------------------------------------------------------------
// Kernel 2: O[m, n] = sum_k E[m, k] * Wm[n, kOff + k]      (M=64, N=256, K=512)
// One wave per 16x16 tile, V_WMMA_F32_16X16X4_F32, K stepped by 4.
// A (16x4 f32) lane layout: lane&15 = M row; VGPR0/1 hold K, K+1; lanes>=16
// hold K+2, K+3.  B (4x16 f32) is the transposed-symmetric layout: lane&15 =
// N column.  Both therefore load with the same per-lane addressing pattern.
// ---------------------------------------------------------------------------
__global__ void PositionTerm_gemm_kernel(const float* __restrict__ E,
                                         const float* __restrict__ Wm,
                                         int kOff,
                                         float* __restrict__ O) {
    const int lane = threadIdx.x;        // wave32: 0..31, EXEC all ones
    const int row  = lane & 15;
    const int hi   = lane >> 4;          // 0 or 1 -> which K pair this lane holds
    const int m0   = blockIdx.x * 16;
    const int n0   = blockIdx.y * 16;

    const float* ea = E  + (size_t)(m0 + row) * EMB_DIM;
    const float* wb = Wm + (size_t)(n0 + row) * (2 * EMB_DIM) + kOff;

    v8f c = {};
#pragma unroll 8
    for (int k0 = 0; k0 < EMB_DIM; k0 += 4) {
        const int kA = k0 + hi * 2;
        v2f a; a.x = ea[kA]; a.y = ea[kA + 1];   // A[m, kA], A[m, kA+1]
        v2f b; b.x = wb[kA]; b.y = wb[kA + 1];   // B[kA, n], B[kA+1, n]
        // D = A x B + C  (8 args: neg_a, A, neg_b, B, c_mod, C, reuse_a, reuse_b)
        c = __builtin_amdgcn_wmma_f32_16x16x4_f32(
                false, a, false, b, (short)0, c, false, false);
    }

    // C/D layout: lane&15 = N column; VGPR r = row M = r + 8*(lane>=16)
    float* o = O + n0 + row;
#pragma unroll
    for (int r = 0; r < 8; ++r) {
        o[(size_t)(m0 + r + hi * 8) * OUT_DIM] = c[r];
    }
}

// ---------------------------------------------------------------------------
// Kernel 3: out[b,h,w,o] = OX[h,o] + OY[w,o]  -- 32 MiB of pure streaming
// stores.  One float4 per thread, non-temporal (write-once, never re-read).
// flat float4 index bits: [b:3][h:6][w:6][o4:6]
// ---------------------------------------------------------------------------
__global__ void PositionTerm_bcast_kernel(const float* __restrict__ OX,
                                          const float* __restrict__ OY,
                                          float* __restrict__ out) {
    const unsigned idx = blockIdx.x * blockDim.x + threadIdx.x; // float4 index
    const unsigned o4 = idx & 63;
    const unsigned w  = (idx >> 6) & 63;
    const unsigned h  = (idx >> 12) & 63;

    const v4f a = ((const v4f*)OX)[h * (OUT_DIM / 4) + o4];
    const v4f b = ((const v4f*)OY)[w * (OUT_DIM / 4) + o4];
    const v4f v = a + b;
    __builtin_nontemporal_store(v, ((v4f*)out) + idx);
}

extern "C" void kernel_launch(void* const* d_in, const int* in_sizes, int n_in,
                              void* d_out, int out_size, void* d_ws, size_t ws_size,
                              hipStream_t stream) {
    const float* W = (const float*)d_in[0];   // (256, 1024) f32
    float* ws = (float*)d_ws;
    float* ex = ws;                                // 64*512
    float* ey = ex + HW * EMB_DIM;                 // 64*512
    float* OX = ey + HW * EMB_DIM;                 // 64*256
    float* OY = OX + HW * OUT_DIM;                 // 64*256
    float* out = (float*)d_out;                    // 8*64*64*256

    // 1) sinusoid tables: 2*64*512 = 65536 elements
    PositionTerm_embed_kernel<<<(2 * HW * EMB_DIM) / 256, 256, 0, stream>>>(ex, ey);

    // 2) two 64x256x512 WMMA GEMMs (one wave = one 16x16 tile)
    dim3 grid(HW / 16, OUT_DIM / 16);
    PositionTerm_gemm_kernel<<<grid, 32, 0, stream>>>(ex, W, 0,       OX);
    PositionTerm_gemm_kernel<<<grid, 32, 0, stream>>>(ey, W, EMB_DIM, OY);

    // 3) broadcast-add: 8*64*64*64 float4 stores = 32 MiB
    const unsigned n4 = BATCH * HW * HW * (OUT_DIM / 4); // 2^21
    PositionTerm_bcast_kernel<<<n4 / 256, 256, 0, stream>>>(OX, OY, out);
}